// GeoFormer_49435073577490
// MI455X (gfx1250) — compile-verified
//
#include <hip/hip_runtime.h>
#include <math.h>

// ---------------------------------------------------------------------------
// GeoFormer post-processing for MI455X (gfx1250, wave32).
//   Q=128 queries, NFG=200000 fg points, NPTS=250000 points, C=20 classes.
// Dominant traffic: mask_logits stream (102.4 MB) + proposals output (128 MB).
// einsum mask_f @ softmax(sem) via V_WMMA_F32_16X16X4_F32 (exact f32 math,
// A-matrix is {0,1}); npoints / mask_score row sums folded into the K-loop
// (sigmoid via native v_rcp_f32, batched once per 16-K trip). K-loop unrolled
// 4x -> 4 back-to-back WMMAs per trip behind an 8-load clause.
// Partial tiles combine via global f32 atomics.
// ---------------------------------------------------------------------------

#define Qn     128
#define NFGn   200000
#define NPTSn  250000
#define Cn     20
#define NPADn  32              // C padded to 2 WMMA N-tiles
#define KBLOCKS 250            // 250 * 800 = 200000
#define KCHUNK  800            // divisible by 16 (4x-unrolled 4-K steps)

typedef __attribute__((ext_vector_type(2))) float v2f;
typedef __attribute__((ext_vector_type(8))) float v8f;

// ---------------- zeroing ----------------

__global__ void k_zero_prop(float* __restrict__ p)
{
    const size_t n4 = (size_t)Qn * NPTSn / 4;        // 8M float4
    float4 z = make_float4(0.f, 0.f, 0.f, 0.f);
    for (size_t i = (size_t)blockIdx.x * blockDim.x + threadIdx.x;
         i < n4; i += (size_t)gridDim.x * blockDim.x)
        ((float4*)p)[i] = z;
}

__global__ void k_zero_acc(float* __restrict__ sem_accum,
                           float* __restrict__ npoints,
                           float* __restrict__ masknum)
{
    for (int i = threadIdx.x; i < Qn * NPADn; i += blockDim.x) sem_accum[i] = 0.f;
    if ((int)threadIdx.x < Qn) { npoints[threadIdx.x] = 0.f; masknum[threadIdx.x] = 0.f; }
}

// ---------------- softmax of semantic scores, written TRANSPOSED+padded -----
// semT[32][NFG]: row c = class channel, rows 20..31 = 0.  Writes are fully
// coalesced (thread n writes semT[c*NFG + n]).

__global__ void k_sem_softmax(const float* __restrict__ sem,   // [NFG, 20]
                              float* __restrict__ semT)        // [32, NFG]
{
    int n = blockIdx.x * blockDim.x + threadIdx.x;
    if (n >= NFGn) return;
    const float* row = sem + (size_t)n * Cn;
    float v[Cn];
    float mx = -1e30f;
#pragma unroll
    for (int c = 0; c < Cn; ++c) { v[c] = row[c]; mx = fmaxf(mx, v[c]); }
    float s = 0.f;
#pragma unroll
    for (int c = 0; c < Cn; ++c) { v[c] = __expf(v[c] - mx); s += v[c]; }
    float inv = 1.f / s;
#pragma unroll
    for (int c = 0; c < Cn; ++c) semT[(size_t)c * NFGn + n] = v[c] * inv;
#pragma unroll
    for (int c = Cn; c < NPADn; ++c) semT[(size_t)c * NFGn + n] = 0.f;
}

// ---------------- per-query class softmax/argmax ----------------

__global__ void k_cls(const float* __restrict__ cls_logits,    // [Q, 20]
                      int* __restrict__ cls_pred,
                      float* __restrict__ cls_score)
{
    int q = threadIdx.x;
    if (q >= Qn) return;
    const float* row = cls_logits + (size_t)q * Cn;
    float mx = row[0]; int am = 0;
#pragma unroll
    for (int c = 1; c < Cn; ++c) { float x = row[c]; if (x > mx) { mx = x; am = c; } }
    float s = 0.f;
#pragma unroll
    for (int c = 0; c < Cn; ++c) s += __expf(row[c] - mx);
    cls_pred[q]  = am;
    cls_score[q] = 1.f / s;    // softmax value at the argmax
}

// ---------------- WMMA GEMM + fused row reductions ----------------
// 512 threads = 16 waves; wave w -> (mtile = w>>1 in [0,8), ntile = w&1).
// A: mask_f tile 16x4 f32 (lane l: row l&15, k-offset (l>>4)*2).
// B: semT tile 4x16 f32 (lane l: col l&15, same k-offset) -> contiguous float2.

__global__ void __launch_bounds__(512)
k_wmma(const float* __restrict__ mask_logits,   // [Q, NFG]
       const float* __restrict__ semT,          // [32, NFG]
       float* __restrict__ sem_accum,           // [Q, 32]
       float* __restrict__ npoints,             // [Q]
       float* __restrict__ masknum)             // [Q]
{
    const int lane  = threadIdx.x & 31;
    const int wave  = threadIdx.x >> 5;
    const int mtile = wave >> 1;
    const int ntile = wave & 1;
    const int l15   = lane & 15;
    const int koff  = (lane >> 4) << 1;          // 0 or 2

    int k0   = blockIdx.x * KCHUNK;
    int kend = k0 + KCHUNK;
    if (kend > NFGn) kend = NFGn;

    const float* __restrict__ arow = mask_logits + (size_t)(mtile * 16 + l15) * NFGn + koff;
    const float* __restrict__ brow = semT        + (size_t)(ntile * 16 + l15) * NFGn + koff;

    v8f acc = {0.f, 0.f, 0.f, 0.f, 0.f, 0.f, 0.f, 0.f};
    float cnt = 0.f, pr = 0.f;

    for (int k = k0; k < kend; k += 16) {
        if ((k & 1023) == 0) {                    // stream-ahead hints
            __builtin_prefetch(arow + k + 4096, 0, 1);
            __builtin_prefetch(brow + k + 4096, 0, 1);
        }

        // phase 1: issue all 8 loads (single clause, max MLP)
        float2 av[4], bv[4];
#pragma unroll
        for (int u = 0; u < 4; ++u) {
            av[u] = *(const float2*)(arow + k + u * 4);
            bv[u] = *(const float2*)(brow + k + u * 4);
        }

        // phase 2: build A operands, 4 back-to-back WMMAs
        v2f a[4];
#pragma unroll
        for (int u = 0; u < 4; ++u) {
            a[u].x = (av[u].x >= 0.f) ? 1.f : 0.f;   // sigmoid(x)>=0.5 <=> x>=0
            a[u].y = (av[u].y >= 0.f) ? 1.f : 0.f;
        }
#pragma unroll
        for (int u = 0; u < 4; ++u) {
            v2f b;
            b.x = bv[u].x;
            b.y = bv[u].y;
            acc = __builtin_amdgcn_wmma_f32_16x16x4_f32(
                      false, a[u], false, b, (short)0, acc, false, false);
        }

        // phase 3: one wave-uniform branch per trip; 8 exps/rcps pipeline
        if (ntile == 0) {
#pragma unroll
            for (int u = 0; u < 4; ++u) {
                float s0 = __builtin_amdgcn_rcpf(1.f + __expf(-av[u].x));
                float s1 = __builtin_amdgcn_rcpf(1.f + __expf(-av[u].y));
                cnt += a[u].x + a[u].y;
                pr  += a[u].x * s0 + a[u].y * s1;
            }
        }
    }

    // accumulate the 16x16 tile: VGPR r holds M = r + 8*(lane>=16), N = lane&15
#pragma unroll
    for (int r = 0; r < 8; ++r) {
        int mm = mtile * 16 + r + ((lane >> 4) << 3);
        int nn = ntile * 16 + l15;
        atomicAdd(&sem_accum[mm * NPADn + nn], acc[r]);
    }

    if (ntile == 0) {
        cnt += __shfl_xor(cnt, 16);
        pr  += __shfl_xor(pr, 16);
        if (lane < 16) {
            atomicAdd(&npoints[mtile * 16 + l15], cnt);
            atomicAdd(&masknum[mtile * 16 + l15], pr);
        }
    }
}

// ---------------- final per-query scoring ----------------

__global__ void k_final(const float* __restrict__ npoints,
                        const float* __restrict__ masknum,
                        const float* __restrict__ sem_accum,
                        const int*   __restrict__ cls_pred,
                        const float* __restrict__ cls_score,
                        float* __restrict__ out_cls,
                        float* __restrict__ out_score,
                        float* __restrict__ out_cond,
                        int*   __restrict__ cond)
{
    int q = threadIdx.x;
    if (q >= Qn) return;
    float np     = npoints[q];
    float denom  = np + 1e-6f;
    float mscore = masknum[q] / denom;
    int   cp     = cls_pred[q];
    float sscore = sem_accum[q * NPADn + cp] / denom;
    float score  = mscore * sqrtf(cls_score[q]) * sscore;
    int cnd = (cp >= 4) && (np >= 100.0f) && (mscore >= 0.5f);
    cond[q]      = cnd;
    out_cls[q]   = cnd ? (float)cp : 0.f;
    out_score[q] = cnd ? score : 0.f;
    out_cond[q]  = cnd ? 1.f : 0.f;
}

// ---------------- scatter selected masks into full point cloud ----------------

__global__ void k_scatter(const float* __restrict__ mask_logits,
                          const int*   __restrict__ fg_idxs,
                          const int*   __restrict__ cond,
                          float* __restrict__ out_prop)       // [Q, NPTS]
{
    int q = blockIdx.y;
    if (!cond[q]) return;                        // uniform early-out per row
    int n = blockIdx.x * blockDim.x + threadIdx.x;
    if (n >= NFGn) return;
    if (mask_logits[(size_t)q * NFGn + n] >= 0.f)
        out_prop[(size_t)q * NPTSn + fg_idxs[n]] = 1.0f;
}

// ---------------- host-side launch ----------------

extern "C" void kernel_launch(void* const* d_in, const int* in_sizes, int n_in,
                              void* d_out, int out_size, void* d_ws, size_t ws_size,
                              hipStream_t stream)
{
    const float* mask_logits = (const float*)d_in[0];   // [1, Q, NFG]
    const float* cls_logits  = (const float*)d_in[1];   // [1, Q, C]
    const int*   fg_idxs     = (const int*)  d_in[2];   // [NFG]
    // d_in[3], d_in[4] : batch offsets (unused, batch 0 spans everything)
    const float* semantic    = (const float*)d_in[5];   // [NFG, C]

    float* ws        = (float*)d_ws;
    float* semT      = ws;                               // 32*NFG floats (25.6 MB)
    float* sem_accum = semT + (size_t)NPADn * NFGn;      // Q*32
    float* npoints   = sem_accum + Qn * NPADn;           // Q
    float* masknum   = npoints + Qn;                     // Q
    float* cls_score = masknum + Qn;                     // Q
    int*   cls_pred  = (int*)(cls_score + Qn);           // Q
    int*   cond      = cls_pred + Qn;                    // Q

    float* out       = (float*)d_out;
    float* out_cls   = out;                              // [Q]
    float* out_score = out + Qn;                         // [Q]
    float* out_prop  = out + 2 * Qn;                     // [Q, NPTS]
    float* out_cond  = out + 2 * Qn + (size_t)Qn * NPTSn;// [Q]

    k_zero_prop<<<8192, 256, 0, stream>>>(out_prop);
    k_zero_acc <<<1, 512, 0, stream>>>(sem_accum, npoints, masknum);
    k_sem_softmax<<<(NFGn + 255) / 256, 256, 0, stream>>>(semantic, semT);
    k_cls<<<1, 128, 0, stream>>>(cls_logits, cls_pred, cls_score);
    k_wmma<<<KBLOCKS, 512, 0, stream>>>(mask_logits, semT, sem_accum, npoints, masknum);
    k_final<<<1, 128, 0, stream>>>(npoints, masknum, sem_accum, cls_pred, cls_score,
                                   out_cls, out_score, out_cond, cond);
    k_scatter<<<dim3((NFGn + 255) / 256, Qn), 256, 0, stream>>>(
        mask_logits, fg_idxs, cond, out_prop);
}